// MultiViewConsistencyLoss_11931419148402
// MI455X (gfx1250) — compile-verified
//
#include <hip/hip_runtime.h>
#include <math.h>

// ---------------------------------------------------------------------------
// MultiViewConsistencyLoss on MI455X (gfx1250, wave32, WMMA)
//
// d2(a,b) = |a|^2 + |b|^2 - 2 a.b  computed as one V_WMMA_F32_16X16X4_F32 per
// 16x16 tile:  A = [-2ax,-2ay,-2az, 1] (16x4), B = [bx;by;bz;|b|^2] (4x16),
// C = row-broadcast of |a|^2.  Running min kept in d^2 space (sqrt monotone),
// sqrt only once per row at the end.
// ---------------------------------------------------------------------------

typedef __attribute__((ext_vector_type(2))) float v2f;
typedef __attribute__((ext_vector_type(8))) float v8f;

#define NPTS   8192
#define NVIEW  4
#define NPAIR  6
#define COL_TILES      (NPTS / 16)   // 512 column tiles per pair
#define WAVES_PER_PAIR (NPTS / 32)   // each wave owns 32 rows (two 16-row tiles)
#define TOTAL_WAVES    (NPAIR * WAVES_PER_PAIR)   // 1536

// ---- 1) rigid transform + squared-norm precompute -------------------------
__global__ __launch_bounds__(256) void mvc_transform_kernel(
    const float* __restrict__ pts,    // [V][N][3]
    const float* __restrict__ poses,  // [V][4][4] row-major
    float4* __restrict__ world)       // [V][N] = (x,y,z,|w|^2)
{
  int idx = blockIdx.x * 256 + threadIdx.x;
  if (idx >= NVIEW * NPTS) return;
  int v = idx >> 13;                      // / NPTS
  const float* P = poses + v * 16;
  const float* p = pts + (size_t)idx * 3;
  float x = p[0], y = p[1], z = p[2];
  float wx = P[0] * x + P[1] * y + P[2]  * z + P[3];
  float wy = P[4] * x + P[5] * y + P[6]  * z + P[7];
  float wz = P[8] * x + P[9] * y + P[10] * z + P[11];
  float nr = wx * wx + wy * wy + wz * wz;
  world[idx] = make_float4(wx, wy, wz, nr);
}

// ---- 2) pairwise chamfer row-min via WMMA ---------------------------------
__global__ __launch_bounds__(256) void mvc_chamfer_kernel(
    const float4* __restrict__ world,
    float* __restrict__ partials)     // [TOTAL_WAVES]
{
  const int lane = threadIdx.x & 31;
  const int wave = threadIdx.x >> 5;
  const int gw   = blockIdx.x * 8 + wave;           // global wave id
  const int pair = gw / WAVES_PER_PAIR;             // 0..5
  const int tp   = gw % WAVES_PER_PAIR;             // 32-row block index

  // pair -> (i, j), enumeration order (0,1)(0,2)(0,3)(1,1)(1,2)(2,3)
  const int i = (pair < 3) ? 0 : ((pair < 5) ? 1 : 2);
  const int j = (pair < 3) ? (pair + 1) : ((pair < 5) ? (pair - 2) : 3);

  const int half = lane >> 4;     // 0: K={0,1}/rows 0-7 of C; 1: K={2,3}/rows 8-15
  const int lm   = lane & 15;     // M (for A) / N (for B) within tile

  // --- A operands and C (row-norm) accumulators for the two row tiles ---
  const float4* aP = world + i * NPTS + tp * 32;
  float4 ap0 = aP[lm];            // row point for tile 0
  float4 ap1 = aP[16 + lm];       // row point for tile 1

  v2f A0, A1;                     // A layout: VGPR0/1 = K 2*half / 2*half+1
  A0.x = half ? (-2.0f * ap0.z) : (-2.0f * ap0.x);
  A0.y = half ? 1.0f             : (-2.0f * ap0.y);
  A1.x = half ? (-2.0f * ap1.z) : (-2.0f * ap1.x);
  A1.y = half ? 1.0f             : (-2.0f * ap1.y);

  v8f C0, C1, acc0, acc1;         // C[v] broadcasts |a_row|^2, row = v + 8*half
#pragma unroll
  for (int v = 0; v < 8; ++v) {
    C0[v] = aP[v + 8 * half].w;
    C1[v] = aP[16 + v + 8 * half].w;
    acc0[v] = 3.4e38f;
    acc1[v] = 3.4e38f;
  }

  // --- sweep all column tiles of view j, one b64 load feeds two WMMAs ---
  // B layout: lane lm, VGPR0/1 = K 2*half / 2*half+1 -> float2 slice of float4
  const char* bp = (const char*)(world + j * NPTS) + lm * 16 + half * 8;
#pragma unroll 4
  for (int ct = 0; ct < COL_TILES; ++ct) {
    v2f B = *(const v2f*)(bp + (size_t)ct * 256);
    v8f D0 = __builtin_amdgcn_wmma_f32_16x16x4_f32(
        false, A0, false, B, (short)0, C0, false, false);
    v8f D1 = __builtin_amdgcn_wmma_f32_16x16x4_f32(
        false, A1, false, B, (short)0, C1, false, false);
#pragma unroll
    for (int v = 0; v < 8; ++v) {
      acc0[v] = fminf(acc0[v], D0[v]);
      acc1[v] = fminf(acc1[v], D1[v]);
    }
  }

  // --- reduce: min over the 16 in-tile columns lives across lanes 0-15 /
  // 16-31 of each C/D VGPR (N = lane%16). Butterfly-min within each half,
  // then sqrt(clamp) per row and sum the 32 row minima.
  float rowsum = 0.0f;
#pragma unroll
  for (int v = 0; v < 8; ++v) {
    float m0 = acc0[v];
    m0 = fminf(m0, __shfl_xor(m0, 1));
    m0 = fminf(m0, __shfl_xor(m0, 2));
    m0 = fminf(m0, __shfl_xor(m0, 4));
    m0 = fminf(m0, __shfl_xor(m0, 8));
    rowsum += sqrtf(fmaxf(m0, 0.0f));
    float m1 = acc1[v];
    m1 = fminf(m1, __shfl_xor(m1, 1));
    m1 = fminf(m1, __shfl_xor(m1, 2));
    m1 = fminf(m1, __shfl_xor(m1, 4));
    m1 = fminf(m1, __shfl_xor(m1, 8));
    rowsum += sqrtf(fmaxf(m1, 0.0f));
  }
  rowsum += __shfl_xor(rowsum, 16);   // combine rows 0-7(+tile) with 8-15(+tile)

  if (lane == 0)
    partials[gw] = rowsum * (1.0f / ((float)NPTS * (float)NPAIR));
}

// ---- 3) deterministic fixed-order reduction -------------------------------
__global__ __launch_bounds__(256) void mvc_reduce_kernel(
    const float* __restrict__ partials, int n, float* __restrict__ out)
{
  __shared__ float s[256];
  float v = 0.0f;
  for (int idx = threadIdx.x; idx < n; idx += 256) v += partials[idx];
  s[threadIdx.x] = v;
  __syncthreads();
  for (int off = 128; off > 0; off >>= 1) {
    if ((int)threadIdx.x < off) s[threadIdx.x] += s[threadIdx.x + off];
    __syncthreads();
  }
  if (threadIdx.x == 0) out[0] = s[0];
}

extern "C" void kernel_launch(void* const* d_in, const int* in_sizes, int n_in,
                              void* d_out, int out_size, void* d_ws, size_t ws_size,
                              hipStream_t stream) {
  const float* pts   = (const float*)d_in[0];   // (4, 8192, 3) f32
  const float* poses = (const float*)d_in[1];   // (4, 4, 4)    f32
  float* out = (float*)d_out;

  float4* world   = (float4*)d_ws;                                      // 512 KB
  float* partials = (float*)((char*)d_ws + (size_t)NVIEW * NPTS * sizeof(float4));

  mvc_transform_kernel<<<(NVIEW * NPTS + 255) / 256, 256, 0, stream>>>(
      pts, poses, world);
  mvc_chamfer_kernel<<<TOTAL_WAVES / 8, 256, 0, stream>>>(world, partials);
  mvc_reduce_kernel<<<1, 256, 0, stream>>>(partials, TOTAL_WAVES, out);
}